// LModel22_26130581028969
// MI455X (gfx1250) — compile-verified
//
#include <hip/hip_runtime.h>
#include <math.h>

#define NVERT 20
#define NSRC 4
#define HWDIM 800
#define EPSF 1e-6f

typedef __attribute__((ext_vector_type(2))) float v2f;
typedef __attribute__((ext_vector_type(8))) float v8f;

// One workgroup (256 threads = 8 wave32) per edge.
__global__ __launch_bounds__(256) void edge_ncc_kernel(
    const float* __restrict__ start_points,  // (E,3)
    const float* __restrict__ end_points,    // (E,3)
    const float* __restrict__ imgs,          // (5,800,800)
    const float* __restrict__ trans,         // (4,3,4)
    const float* __restrict__ K,             // (3,3)
    float* __restrict__ out,                 // loss (4,E) then mask (4,E)
    int E)
{
    const int e = blockIdx.x;
    if (e >= E) return;
    const int tid  = threadIdx.x;
    const int lane = tid & 31;
    const int wave = tid >> 5;

    // ---------------- per-edge setup (redundant per thread; L2-resident) ----
    const float sx = start_points[e * 3 + 0];
    const float sy = start_points[e * 3 + 1];
    const float sz = start_points[e * 3 + 2];
    const float ex = end_points[e * 3 + 0];
    const float ey = end_points[e * 3 + 1];
    const float ez = end_points[e * 3 + 2];

    const float dl0 = ex - sx, dl1 = ey - sy, dl2 = ez - sz;
    const float len = sqrtf(dl0 * dl0 + dl1 * dl1 + dl2 * dl2);
    int nh = (int)(len / 0.01f);
    nh = nh < 2 ? 2 : (nh > 1000 ? 1000 : nh);
    const int total = nh * NVERT;
    const float cnt = (float)total;
    const float inv_nh1 = 1.0f / (float)(nh - 1);

    const float k00 = K[0], k01 = K[1], k02 = K[2];
    const float k10 = K[3], k11 = K[4], k12 = K[5];
    const float k20 = K[6], k21 = K[7], k22 = K[8];

    // 5 cameras: projected endpoint p0, direction d, perpendicular offset u
    float P0x[5], P0y[5], Dx[5], Dy[5], Ux[5], Uy[5];
#pragma unroll
    for (int c = 0; c < 5; ++c) {
        float m[12];
        if (c == 0) {
            m[0] = k00; m[1] = k01; m[2]  = k02; m[3]  = 0.f;
            m[4] = k10; m[5] = k11; m[6]  = k12; m[7]  = 0.f;
            m[8] = k20; m[9] = k21; m[10] = k22; m[11] = 0.f;
        } else {
            const float* T = trans + (c - 1) * 12;
#pragma unroll
            for (int t = 0; t < 12; ++t) m[t] = T[t];
        }
        const float az = m[8] * sx + m[9] * sy + m[10] * sz + m[11];
        const float ax = (m[0] * sx + m[1] * sy + m[2] * sz + m[3]) / (az + EPSF);
        const float ay = (m[4] * sx + m[5] * sy + m[6] * sz + m[7]) / (az + EPSF);
        const float bz = m[8] * ex + m[9] * ey + m[10] * ez + m[11];
        const float bx = (m[0] * ex + m[1] * ey + m[2] * ez + m[3]) / (bz + EPSF);
        const float by = (m[4] * ex + m[5] * ey + m[6] * ez + m[7]) / (bz + EPSF);
        const float dx = bx - ax, dy = by - ay;
        const float nrm = sqrtf(dx * dx + dy * dy);
        const float sc = 0.00167f / (nrm + EPSF);
        P0x[c] = ax; P0y[c] = ay;
        Dx[c] = dx;  Dy[c] = dy;
        Ux[c] = dy * sc;  Uy[c] = -dx * sc;   // cross([dx,dy,0],[0,0,1]) = (dy,-dx)
    }

    // acc: 0=sum1, 1=sum11, then per src s: 2+3s=sum2, 3+3s=sum22, 4+3s=sum12
    float acc[14];
#pragma unroll
    for (int q = 0; q < 14; ++q) acc[q] = 0.f;
    unsigned vmask = 0xFu;

    // ---------------- sample loop -------------------------------------------
    for (int idx = tid; idx < total; idx += 256) {
        const int i = idx / NVERT;
        const int j = idx - i * NVERT;
        const float cx = (float)i * inv_nh1;
        const float cy = (j < 10) ? (float)(j - 9) * (1.f / 9.f)
                                  : (float)(j - 10) * (1.f / 9.f);
        float vals[5];
        bool  vld[5];
#pragma unroll
        for (int c = 0; c < 5; ++c) {
            float px = Dx[c] * cx + Ux[c] * cy + P0x[c];
            float py = Dy[c] * cx + Uy[c] * cy + P0y[c];
            vld[c] = (px > 0.f) & (px < 1.f) & (py > 0.f) & (py < 1.f);
            px = fminf(fmaxf(px, 0.f), 0.999999f);
            py = fminf(fmaxf(py, 0.f), 0.999999f);
            const float x = px * (float)(HWDIM - 1);
            const float y = py * (float)(HWDIM - 1);
            float fx0 = fminf(fmaxf(floorf(x), 0.f), (float)(HWDIM - 2));
            float fy0 = fminf(fmaxf(floorf(y), 0.f), (float)(HWDIM - 2));
            const int x0 = (int)fx0, y0 = (int)fy0;
            const float wx = x - fx0, wy = y - fy0;
            const float* img = imgs + (size_t)c * HWDIM * HWDIM;
            const size_t b = (size_t)y0 * HWDIM + (size_t)x0;
            const float v00 = img[b];
            const float v01 = img[b + 1];
            const float v10 = img[b + HWDIM];
            const float v11 = img[b + HWDIM + 1];
            vals[c] = (v00 * (1.f - wx) + v01 * wx) * (1.f - wy) +
                      (v10 * (1.f - wx) + v11 * wx) * wy;
        }
        const float s1v = vals[0];
        acc[0] += s1v;
        acc[1] += s1v * s1v;
#pragma unroll
        for (int s = 0; s < NSRC; ++s) {
            const float s2v = vals[1 + s];
            acc[2 + 3 * s] += s2v;
            acc[3 + 3 * s] += s2v * s2v;
            acc[4 + 3 * s] += s1v * s2v;
            if (!(vld[0] && vld[1 + s])) vmask &= ~(1u << s);
        }
    }

    // ---------------- intra-wave reduction (wave32 shuffles) ----------------
#pragma unroll
    for (int q = 0; q < 14; ++q) {
#pragma unroll
        for (int off = 16; off > 0; off >>= 1)
            acc[q] += __shfl_down(acc[q], off, 32);
    }
#pragma unroll
    for (int off = 16; off > 0; off >>= 1)
        vmask &= __shfl_down(vmask, off, 32);

    __shared__ float    wsum[8][16];
    __shared__ unsigned wmask8[8];
    __shared__ float    totals[16];
    __shared__ unsigned maskTot;

    if (lane == 0) {
#pragma unroll
        for (int q = 0; q < 14; ++q) wsum[wave][q] = acc[q];
        wsum[wave][14] = 0.f;
        wsum[wave][15] = 0.f;
        wmask8[wave] = vmask;
    }
    __syncthreads();

    // ---------------- cross-wave reduction via V_WMMA_F32_16X16X4_F32 -------
    // A layout (16x4 f32): lane L<16 -> row M=L, K=0..1 in a[0..1];
    //                      lane L+16 -> row M=L, K=2..3 in a[0..1].
    // With B = ones, D[m][*] = sum_k A[m][k]: one WMMA sums 4 wave-partials
    // per quantity; two chained WMMAs (D fed back as C) sum all 8 waves.
    if (wave == 0) {
        const int q = lane & 15;
        const int b = (lane < 16) ? 0 : 2;
        v2f a1, a2, ones;
        a1[0] = wsum[b][q];     a1[1] = wsum[b + 1][q];
        a2[0] = wsum[b + 4][q]; a2[1] = wsum[b + 5][q];
        ones[0] = 1.f; ones[1] = 1.f;
        v8f cz = {};
        v8f d = __builtin_amdgcn_wmma_f32_16x16x4_f32(
            false, a1, false, ones, (short)0, cz, false, false);
        d = __builtin_amdgcn_wmma_f32_16x16x4_f32(
            false, a2, false, ones, (short)0, d, false, false);
        // D layout: VGPR r, lanes 0-15 -> M=r; lanes 16-31 -> M=r+8. N=0 column.
        if (lane == 0) {
#pragma unroll
            for (int r = 0; r < 8; ++r) totals[r] = d[r];
            unsigned mt = wmask8[0];
#pragma unroll
            for (int w = 1; w < 8; ++w) mt &= wmask8[w];
            maskTot = mt;
        } else if (lane == 16) {
#pragma unroll
            for (int r = 0; r < 8; ++r) totals[8 + r] = d[r];
        }
    }
    __syncthreads();

    // ---------------- finalize NCC loss + mask ------------------------------
    if (tid == 0) {
        const float sum1  = totals[0];
        const float sum11 = totals[1];
        const float var1  = sum11 - sum1 * sum1 / cnt;
        const unsigned mt = maskTot;
#pragma unroll
        for (int s = 0; s < NSRC; ++s) {
            const float s2  = totals[2 + 3 * s];
            const float s22 = totals[3 + 3 * s];
            const float s12 = totals[4 + 3 * s];
            const float cov  = s12 - sum1 * s2 / cnt;
            const float var2 = s22 - s2 * s2 / cnt;
            const float loss = 1.f - cov / sqrtf(var1 * var2 + EPSF);
            out[s * E + e] = loss;
            out[NSRC * E + s * E + e] = ((mt >> s) & 1u) ? 1.f : 0.f;
        }
    }
}

extern "C" void kernel_launch(void* const* d_in, const int* in_sizes, int n_in,
                              void* d_out, int out_size, void* d_ws, size_t ws_size,
                              hipStream_t stream)
{
    const float* start = (const float*)d_in[0];
    const float* end   = (const float*)d_in[1];
    const float* imgs  = (const float*)d_in[2];
    const float* trans = (const float*)d_in[3];
    const float* K     = (const float*)d_in[4];
    const int E = in_sizes[0] / 3;   // (E,3) start points
    edge_ncc_kernel<<<E, 256, 0, stream>>>(start, end, imgs, trans, K,
                                           (float*)d_out, E);
}